// Augmentor_54597624267034
// MI455X (gfx1250) — compile-verified
//
#include <hip/hip_runtime.h>
#include <math.h>

typedef float v2f __attribute__((ext_vector_type(2)));
typedef float v8f __attribute__((ext_vector_type(8)));

#define MAX_LOGSTD 10.0f

// ---------------- degree / normalization ----------------
__global__ void k_deg_init(float* deg, int n) {
    int i = blockIdx.x * blockDim.x + threadIdx.x;
    if (i < n) deg[i] = 1.0f;   // self-loop contributes 1
}

__global__ void k_deg_accum(const int* __restrict__ dst, float* deg, int E) {
    int e = blockIdx.x * blockDim.x + threadIdx.x;
    if (e < E) unsafeAtomicAdd(&deg[dst[e]], 1.0f);
}

__global__ void k_dinv(float* d, int n) {
    int i = blockIdx.x * blockDim.x + threadIdx.x;
    if (i < n) {
        float v = d[i];
        d[i] = v > 0.0f ? rsqrtf(v) : 0.0f;
    }
}

// ---------------- weight transposes into WMMA-fragment-ready packed form ---------
// Bp is [(K/2), Ncol] of float2: Bp[kp*Ncol + n] = { W[n][2kp], W[n][2kp+1] }
// so a B fragment (K rows kk,kk+1 ; col n) is ONE b64 load per lane.
__global__ void k_transpose_pack(const float* __restrict__ W, float2* __restrict__ Bp,
                                 int Ncol /*out feats*/, int K) {
    int i = blockIdx.x * blockDim.x + threadIdx.x;
    int total = (K >> 1) * Ncol;
    if (i < total) {
        int kp = i / Ncol;
        int n  = i - kp * Ncol;
        const float* w = W + (size_t)n * K + 2 * kp;
        Bp[i] = make_float2(w[0], w[1]);
    }
}

__global__ void k_transpose_pack2(const float* __restrict__ Wmu, const float* __restrict__ Wls,
                                  float2* __restrict__ Bp, int H, int D) {
    // combined output width 2D ; column j<D -> Wmu row j, else Wls row j-D
    int i = blockIdx.x * blockDim.x + threadIdx.x;
    int Ncol = 2 * D;
    int total = (H >> 1) * Ncol;
    if (i < total) {
        int kp = i / Ncol;
        int j  = i - kp * Ncol;
        const float* w = (j < D) ? (Wmu + (size_t)j * H) : (Wls + (size_t)(j - D) * H);
        Bp[i] = make_float2(w[2 * kp], w[2 * kp + 1]);
    }
}

// ---------------- fp32 WMMA GEMM: C[M,Ncol] = A[M,K] * B[K,Ncol] ----------------
// Block = 128 threads = 4 waves; wave computes a 16(M) x 64(N) strip via 4
// V_WMMA_F32_16X16X4_F32 accumulators. Full fp32 precision (GEMM is ~33 GFLOP
// total -- far from the roofline bottleneck -- so keep exact fp32 semantics).
// B comes pre-packed as float2 K-row pairs: one b64 load per fragment.
__global__ __launch_bounds__(128) void k_gemm_wmma(const float* __restrict__ A,
                                                   const float2* __restrict__ Bp,
                                                   float* __restrict__ C,
                                                   int M, int K, int Ncol) {
    const int lane = threadIdx.x & 31;
    const int wave = threadIdx.x >> 5;
    const int half = lane >> 4;      // 0: K pair {0,1}; 1: K pair {2,3}
    const int l    = lane & 15;      // M (for A) / N (for B,C) within tile
    const int row0 = blockIdx.y * 64 + wave * 16;
    const int col0 = blockIdx.x * 64;

    int r = row0 + l;
    if (r >= M) r = M - 1;           // clamp (stores are guarded)
    const float* Ar = A + (size_t)r * K;

    v8f c0 = {}, c1 = {}, c2 = {}, c3 = {};

    for (int k = 0; k < K; k += 4) {
        const int kk = k + 2 * half;
        // A 16x4 f32 fragment: VGPR0=K(kk), VGPR1=K(kk+1)  -> one b64 load
        v2f a = *(const v2f*)(Ar + kk);
        // B 4x16 f32 fragments: packed {row kk, row kk+1} float2 per column
        const float2* bp = Bp + (size_t)(kk >> 1) * Ncol + col0 + l;
        float2 f0 = bp[0], f1 = bp[16], f2 = bp[32], f3 = bp[48];
        v2f b0 = (v2f){f0.x, f0.y};
        v2f b1 = (v2f){f1.x, f1.y};
        v2f b2 = (v2f){f2.x, f2.y};
        v2f b3 = (v2f){f3.x, f3.y};
        c0 = __builtin_amdgcn_wmma_f32_16x16x4_f32(false, a, false, b0, (short)0, c0, false, false);
        c1 = __builtin_amdgcn_wmma_f32_16x16x4_f32(false, a, false, b1, (short)0, c1, false, false);
        c2 = __builtin_amdgcn_wmma_f32_16x16x4_f32(false, a, false, b2, (short)0, c2, false, false);
        c3 = __builtin_amdgcn_wmma_f32_16x16x4_f32(false, a, false, b3, (short)0, c3, false, false);
    }

    // C layout: VGPR i -> row row0 + i + 8*half, col = col0 + 16t + l
    const int mr = row0 + 8 * half;
#pragma unroll
    for (int i = 0; i < 8; ++i) {
        int rr = mr + i;
        if (rr < M) {
            float* Cp = C + (size_t)rr * Ncol + col0 + l;
            Cp[0]  = c0[i];
            Cp[16] = c1[i];
            Cp[32] = c2[i];
            Cp[48] = c3[i];
        }
    }
}

// ---------------- aggregation ----------------
// agg[i,:] = pre[i,:] * dinv[i]^2   (self-loop term; also initializes buffer)
__global__ void k_selfloop_init(const float* __restrict__ pre, const float* __restrict__ dinv,
                                float* __restrict__ agg, int n, int F) {
    long i = (long)blockIdx.x * blockDim.x + threadIdx.x;
    long total = (long)n * (F >> 2);
    if (i >= total) return;
    int node = (int)(i / (F >> 2));
    float s = dinv[node];
    s = s * s;
    float4 v = ((const float4*)pre)[i];
    v.x *= s; v.y *= s; v.z *= s; v.w *= s;
    ((float4*)agg)[i] = v;
}

// agg[dst,:] += pre[src,:] * dinv[src]*dinv[dst]   (L2-resident f32 atomics:
// the 102MB/51MB accumulators fit in the 192MB L2, so atomics resolve there)
__global__ void k_aggregate(const float* __restrict__ pre, float* agg,
                            const int* __restrict__ src, const int* __restrict__ dst,
                            const float* __restrict__ dinv, int E, int F, int shift) {
    long gid = (long)blockIdx.x * blockDim.x + threadIdx.x;
    int e = (int)(gid >> shift);
    if (e >= E) return;
    int f4 = ((int)gid) & ((1 << shift) - 1);
    int s = src[e], d = dst[e];
    float nrm = dinv[s] * dinv[d];
    float4 m = ((const float4*)(pre + (size_t)s * F))[f4];
    float* ap = agg + (size_t)d * F + 4 * f4;
    unsafeAtomicAdd(ap + 0, m.x * nrm);
    unsafeAtomicAdd(ap + 1, m.y * nrm);
    unsafeAtomicAdd(ap + 2, m.z * nrm);
    unsafeAtomicAdd(ap + 3, m.w * nrm);
}

// h = relu(agg + b), in place
__global__ void k_bias_relu(float* __restrict__ h, const float* __restrict__ b, int n, int F) {
    long i = (long)blockIdx.x * blockDim.x + threadIdx.x;
    long total = (long)n * (F >> 2);
    if (i >= total) return;
    int c4 = (int)(i % (F >> 2));
    float4 bb = ((const float4*)b)[c4];
    float4 v = ((float4*)h)[i];
    v.x = fmaxf(v.x + bb.x, 0.0f);
    v.y = fmaxf(v.y + bb.y, 0.0f);
    v.z = fmaxf(v.z + bb.z, 0.0f);
    v.w = fmaxf(v.w + bb.w, 0.0f);
    ((float4*)h)[i] = v;
}

// z = (agg_mu + b_mu) + eps * exp(min(agg_ls + b_ls, 10))
__global__ void k_finalize(const float* __restrict__ agg2, const float* __restrict__ bmu,
                           const float* __restrict__ bls, const float* __restrict__ eps,
                           float* __restrict__ z, int n, int D) {
    long i = (long)blockIdx.x * blockDim.x + threadIdx.x;
    long total = (long)n * D;
    if (i >= total) return;
    int node = (int)(i / D);
    int d = (int)(i - (long)node * D);
    const float* row = agg2 + (size_t)node * (2 * D);
    float mu = row[d] + bmu[d];
    float ls = fminf(row[D + d] + bls[d], MAX_LOGSTD);
    z[i] = fmaf(eps[i], expf(ls), mu);
}

// s[e] = sigmoid(dot(z[a], z[b])) ; one wave32 per edge, butterfly reduce
__global__ void k_edge_score(const float* __restrict__ z, const int* __restrict__ ta,
                             const int* __restrict__ tb, float* __restrict__ s,
                             int Et, int D) {
    int lane = threadIdx.x & 31;
    int e = blockIdx.x * (blockDim.x >> 5) + (threadIdx.x >> 5);
    if (e >= Et) return;
    int a = ta[e], b = tb[e];
    float2 za = ((const float2*)(z + (size_t)a * D))[lane];
    float2 zb = ((const float2*)(z + (size_t)b * D))[lane];
    float p = za.x * zb.x + za.y * zb.y;
#pragma unroll
    for (int off = 16; off > 0; off >>= 1) p += __shfl_xor(p, off, 32);
    if (lane == 0) s[e] = 1.0f / (1.0f + expf(-p));
}

// ---------------- host launcher ----------------
extern "C" void kernel_launch(void* const* d_in, const int* in_sizes, int n_in,
                              void* d_out, int out_size, void* d_ws, size_t ws_size,
                              hipStream_t stream) {
    const float* x   = (const float*)d_in[0];
    const int*   ei  = (const int*)d_in[1];
    const int*   ti  = (const int*)d_in[2];
    const float* W1  = (const float*)d_in[3];
    const float* b1  = (const float*)d_in[4];
    const float* Wmu = (const float*)d_in[5];
    const float* bmu = (const float*)d_in[6];
    const float* Wls = (const float*)d_in[7];
    const float* bls = (const float*)d_in[8];
    const float* eps = (const float*)d_in[9];

    const int F = 512, H = 256, D = 64;
    const int N  = in_sizes[0] / F;
    const int E  = in_sizes[1] / 2;
    const int Et = in_sizes[2] / 2;
    const int* src = ei;
    const int* dst = ei + E;
    const int* ta  = ti;
    const int* tb  = ti + Et;

    // workspace carve (256B aligned)
    char* base = (char*)d_ws;
    size_t off = 0;
    auto carve = [&](size_t bytes) -> char* {
        off = (off + 255) & ~(size_t)255;
        char* p = base + off;
        off += bytes;
        return p;
    };
    float*  dinv = (float*)carve((size_t)N * 4);          // deg -> dinv (in place)
    float2* BpW1 = (float2*)carve((size_t)F * H * 4);     // packed [(F/2),H] float2
    float2* Bp2  = (float2*)carve((size_t)H * 2 * D * 4); // packed [(H/2),2D] float2
    float*  hw   = (float*)carve((size_t)N * H * 4);      // x@W1^T  [N,256]
    float*  agg1 = (float*)carve((size_t)N * H * 4);      // agg1 -> h (in place)
    // after aggregation-1 + bias_relu, hw region is dead: recycle it
    float*  pre2 = hw;                                    // [N,128] = h@[Wmu|Wls]^T
    float*  agg2 = hw + (size_t)N * 2 * D;                // [N,128]

    const int T = 256;

    // 1) GCN norm
    k_deg_init<<<(N + T - 1) / T, T, 0, stream>>>(dinv, N);
    k_deg_accum<<<(E + T - 1) / T, T, 0, stream>>>(dst, dinv, E);
    k_dinv<<<(N + T - 1) / T, T, 0, stream>>>(dinv, N);

    // 2) layer 1: GEMM (fp32 WMMA) + aggregate + bias/relu
    k_transpose_pack<<<((F / 2) * H + T - 1) / T, T, 0, stream>>>(W1, BpW1, H, F);
    dim3 g1(H / 64, (N + 63) / 64);
    k_gemm_wmma<<<g1, 128, 0, stream>>>(x, BpW1, hw, N, F, H);
    long tot1 = (long)N * (H >> 2);
    k_selfloop_init<<<(unsigned)((tot1 + T - 1) / T), T, 0, stream>>>(hw, dinv, agg1, N, H);
    long tota1 = (long)E * (H >> 2);
    k_aggregate<<<(unsigned)((tota1 + T - 1) / T), T, 0, stream>>>(hw, agg1, src, dst, dinv, E, H, 6);
    k_bias_relu<<<(unsigned)((tot1 + T - 1) / T), T, 0, stream>>>(agg1, b1, N, H);

    // 3) layers 2+3 fused: GEMM [N,256]x[256,128] + aggregate
    k_transpose_pack2<<<((H / 2) * 2 * D + T - 1) / T, T, 0, stream>>>(Wmu, Wls, Bp2, H, D);
    dim3 g2((2 * D) / 64, (N + 63) / 64);
    k_gemm_wmma<<<g2, 128, 0, stream>>>(agg1, Bp2, pre2, N, H, 2 * D);
    long tot2 = (long)N * ((2 * D) >> 2);
    k_selfloop_init<<<(unsigned)((tot2 + T - 1) / T), T, 0, stream>>>(pre2, dinv, agg2, N, 2 * D);
    long tota2 = (long)E * ((2 * D) >> 2);
    k_aggregate<<<(unsigned)((tota2 + T - 1) / T), T, 0, stream>>>(pre2, agg2, src, dst, dinv, E, 2 * D, 5);

    // 4) reparametrize -> z, then edge scoring -> s
    float* zout = (float*)d_out;
    long totz = (long)N * D;
    k_finalize<<<(unsigned)((totz + T - 1) / T), T, 0, stream>>>(agg2, bmu, bls, eps, zout, N, D);
    k_edge_score<<<(Et + (T / 32) - 1) / (T / 32), T, 0, stream>>>(zout, ta, tb,
                                                                  zout + (size_t)N * D, Et, D);
}